// PointNetExtractor_45466523795796
// MI455X (gfx1250) — compile-verified
//
#include <hip/hip_runtime.h>
#include <cstdint>

#define BPC 8

typedef __attribute__((ext_vector_type(2))) float v2f;
typedef __attribute__((ext_vector_type(8))) float v8f;

#if defined(__has_builtin)
#if __has_builtin(__builtin_amdgcn_global_load_async_to_lds_b32)
#define SA_ASYNC 1
#endif
#endif
#ifndef SA_ASYNC
#define SA_ASYNC 0
#endif

// ---------------------------------------------------------------------------
// Build xyz (B,N,3) and feats (B,4,N) incl. box mask channel
// ---------------------------------------------------------------------------
__global__ void build_inputs_kernel(const float* __restrict__ pc,
                                    const float* __restrict__ ctr,
                                    const float* __restrict__ szr,
                                    float* __restrict__ xyz,
                                    float* __restrict__ feats,
                                    int N) {
  int t = blockIdx.x * blockDim.x + threadIdx.x;
  if (t >= BPC * N) return;
  int b = t / N, n = t % N;
  const float* p = pc + (size_t)t * 6;
  float x = p[0], y = p[1], z = p[2];
  xyz[(size_t)t * 3 + 0] = x;
  xyz[(size_t)t * 3 + 1] = y;
  xyz[(size_t)t * 3 + 2] = z;
  float* fb = feats + (size_t)b * 4 * N;
  fb[(size_t)0 * N + n] = p[3];
  fb[(size_t)1 * N + n] = p[4];
  fb[(size_t)2 * N + n] = p[5];
  float c0 = ctr[b * 3 + 0], c1 = ctr[b * 3 + 1], c2 = ctr[b * 3 + 2];
  float h0 = 0.5f * szr[b * 3 + 0], h1 = 0.5f * szr[b * 3 + 1], h2 = 0.5f * szr[b * 3 + 2];
  bool m = (x >= c0 - h0) && (x <= c0 + h0) &&
           (y >= c1 - h1) && (y <= c1 + h1) &&
           (z >= c2 - h2) && (z <= c2 + h2);
  fb[(size_t)3 * N + n] = m ? 1.0f : 0.0f;
}

// ---------------------------------------------------------------------------
// Pad layer-0 weights: (Co,Ci) -> (Co,Cip) with zero-filled K tail
// ---------------------------------------------------------------------------
__global__ void pad_w_kernel(const float* __restrict__ src, float* __restrict__ dst,
                             int Co, int Ci, int Cip) {
  int t = blockIdx.x * blockDim.x + threadIdx.x;
  if (t >= Co * Cip) return;
  int r = t / Cip, c = t % Cip;
  dst[t] = (c < Ci) ? src[(size_t)r * Ci + c] : 0.0f;
}

// ---------------------------------------------------------------------------
// Farthest point sampling: one block per batch. Point coords cached in regs.
// Tie-break = lowest index (matches jnp.argmax first-occurrence).
// ---------------------------------------------------------------------------
template <int PPT>
__global__ void __launch_bounds__(1024)
fps_kernel(const float* __restrict__ xyz, int* __restrict__ outIdx, int N, int S) {
  const int b = blockIdx.x;
  const int tid = threadIdx.x;
  const int T = blockDim.x;
  const float* xb = xyz + (size_t)b * N * 3;

  float px[PPT], py[PPT], pz[PPT], dl[PPT];
#pragma unroll
  for (int j = 0; j < PPT; ++j) {
    int p = tid + j * T;
    px[j] = xb[(size_t)p * 3 + 0];
    py[j] = xb[(size_t)p * 3 + 1];
    pz[j] = xb[(size_t)p * 3 + 2];
    dl[j] = 1e10f;
  }

  __shared__ float swv[32];
  __shared__ int swi[32];
  __shared__ int sfar;

  int far = 0;
  if (tid == 0) outIdx[(size_t)b * S] = 0;

  for (int it = 1; it < S; ++it) {
    float cx = xb[(size_t)far * 3 + 0];
    float cy = xb[(size_t)far * 3 + 1];
    float cz = xb[(size_t)far * 3 + 2];
    float bv = -1.0f;
    int bi = 0;
#pragma unroll
    for (int j = 0; j < PPT; ++j) {
      float dx = px[j] - cx, dy = py[j] - cy, dz = pz[j] - cz;
      float d = dx * dx + dy * dy + dz * dz;
      if (d < dl[j]) dl[j] = d;
      if (dl[j] > bv) { bv = dl[j]; bi = tid + j * T; }
    }
    // wave32 argmax reduce
#pragma unroll
    for (int off = 16; off > 0; off >>= 1) {
      float ov = __shfl_down(bv, off, 32);
      int oi = __shfl_down(bi, off, 32);
      if (ov > bv || (ov == bv && oi < bi)) { bv = ov; bi = oi; }
    }
    int w = tid >> 5, ln = tid & 31;
    if (ln == 0) { swv[w] = bv; swi[w] = bi; }
    __syncthreads();
    if (tid == 0) {
      int nw = T >> 5;
      float v = swv[0];
      int ii = swi[0];
      for (int k = 1; k < nw; ++k)
        if (swv[k] > v || (swv[k] == v && swi[k] < ii)) { v = swv[k]; ii = swi[k]; }
      sfar = ii;
      outIdx[(size_t)b * S + it] = ii;
    }
    __syncthreads();
    far = sfar;
  }
}

// ---------------------------------------------------------------------------
// Gather centroid coordinates: out (B,S,3)
// ---------------------------------------------------------------------------
__global__ void gather_xyz_kernel(const float* __restrict__ xyz, const int* __restrict__ idx,
                                  float* __restrict__ out, int N, int S) {
  int t = blockIdx.x * blockDim.x + threadIdx.x;
  if (t >= BPC * S * 3) return;
  int b = t / (S * 3);
  int rem = t % (S * 3);
  int s = rem / 3, c = rem % 3;
  out[t] = xyz[((size_t)b * N + idx[(size_t)b * S + s]) * 3 + c];
}

// ---------------------------------------------------------------------------
// Ball query: one wave per (b,s). Collect first `ns` in-radius indices in
// ascending order via ballot + prefix popcount; pad with first hit (or 0).
// ---------------------------------------------------------------------------
__global__ void ball_query_kernel(const float* __restrict__ xyz, const float* __restrict__ nx,
                                  int* __restrict__ out, int N, int S, int ns, float r2) {
  int lane = threadIdx.x & 31;
  int q = blockIdx.x * (blockDim.x >> 5) + (threadIdx.x >> 5);
  if (q >= BPC * S) return;
  int b = q / S;
  const float* xb = xyz + (size_t)b * N * 3;
  float qx = nx[(size_t)q * 3 + 0];
  float qy = nx[(size_t)q * 3 + 1];
  float qz = nx[(size_t)q * 3 + 2];
  int* ob = out + (size_t)q * ns;

  int cnt = 0, first = 0;
  bool haveFirst = false;
  for (int base = 0; base < N && cnt < ns; base += 32) {
    int p = base + lane;
    float dx = xb[(size_t)p * 3 + 0] - qx;
    float dy = xb[(size_t)p * 3 + 1] - qy;
    float dz = xb[(size_t)p * 3 + 2] - qz;
    bool pred = (dx * dx + dy * dy + dz * dz) < r2;
    unsigned m = (unsigned)__ballot(pred);
    if (!haveFirst && m != 0u) {
      first = base + (__ffs(m) - 1);
      haveFirst = true;
    }
    int pos = cnt + __popc(m & ((1u << lane) - 1u));
    if (pred && pos < ns) ob[pos] = p;
    cnt += __popc(m);
  }
  int start = cnt < ns ? cnt : ns;
  for (int i = start + lane; i < ns; i += 32) ob[i] = haveFirst ? first : 0;
}

// ---------------------------------------------------------------------------
// Fused SA branch: one 128-thread block per (b, s).
//   - gather 16 neighbor columns into LDS (xyz - centroid manual; feature
//     rows via async global->LDS DMA when available)
//   - 3 MLP layers as fp32 WMMA 16x16x4 tile GEMMs; layer-0 weights pre-padded
//     so the inner loop is branch-free (A frag = one aligned b64 load)
//   - gamma*x+beta + ReLU; last layer max-pooled over neighbors via shfl_xor
// ball==nullptr => identity grouping; nxyz==nullptr => no centroid subtract
// ---------------------------------------------------------------------------
__global__ void __launch_bounds__(128)
sa_branch_kernel(const float* __restrict__ xyzp, const float* __restrict__ featsp,
                 const float* __restrict__ nxyz, const int* __restrict__ ball,
                 float* __restrict__ fout,
                 int Npts, int Cf, int S, int ns, int oOff, int CoutTotal,
                 const float* __restrict__ W0p, const float* __restrict__ g0,
                 const float* __restrict__ bb0, int Co0,
                 const float* __restrict__ W1, const float* __restrict__ g1,
                 const float* __restrict__ bb1, int Co1,
                 const float* __restrict__ W2, const float* __restrict__ g2,
                 const float* __restrict__ bb2, int Co2) {
  const int tid = threadIdx.x;
  const int lane = tid & 31;
  const int wave = tid >> 5;
  const int b = blockIdx.x / S;
  const int s = blockIdx.x % S;
  const int Cin = Cf + 3;
  const int CinP = (Cin + 3) & ~3;

  extern __shared__ float smem[];
  const int szA = ((CinP > Co1) ? CinP : Co1) * 16;
  const int szB = Co0 * 16;
  float* bufA = smem;                // layer0 input / layer1 output (layer2 input)
  float* bufB = smem + szA;          // layer0 output (layer1 input)
  float* maxbuf = smem + szA + szB;  // running max over neighbors, Co2 floats
  __shared__ int sidx[16];
  __shared__ float scent[3];

  for (int c = tid; c < Co2; c += 128) maxbuf[c] = 0.0f;  // ReLU outputs >= 0
  __syncthreads();

  const float* Ws[3] = {W0p, W1, W2};
  const float* Gs[3] = {g0, g1, g2};
  const float* Bs[3] = {bb0, bb1, bb2};
  const int Ks[3] = {CinP, Co0, Co1};  // K extent == row stride (all mult. of 4)
  const int Cos[3] = {Co0, Co1, Co2};

  const int nchunks = ns >> 4;
  for (int ch = 0; ch < nchunks; ++ch) {
    // stage neighbor indices + centroid
    if (tid < 16) {
      int n = ch * 16 + tid;
      sidx[tid] = ball ? ball[(size_t)blockIdx.x * ns + n] : n;
    }
    if (tid < 3) scent[tid] = nxyz ? nxyz[(size_t)blockIdx.x * 3 + tid] : 0.0f;
    __syncthreads();

    // build layer-0 input tile (CinP x 16), zero-padded K rows
    for (int e = tid; e < CinP * 16; e += 128) {
      int c = e >> 4, j = e & 15;
      if (c < 3) {
        int pt = sidx[j];
        bufA[e] = xyzp[((size_t)b * Npts + pt) * 3 + c] - scent[c];
      } else if (c >= Cin) {
        bufA[e] = 0.0f;
      } else {
        const float* src = featsp + ((size_t)b * Cf + (c - 3)) * Npts + sidx[j];
#if SA_ASYNC
        __builtin_amdgcn_global_load_async_to_lds_b32(
            (__attribute__((address_space(1))) int*)(uintptr_t)src,
            (__attribute__((address_space(3))) int*)(uint32_t)(uintptr_t)(bufA + e),
            0, 0);
#else
        bufA[e] = *src;
#endif
      }
    }
#if SA_ASYNC
    asm volatile("s_wait_asynccnt 0x0" ::: "memory");
#endif
    __syncthreads();

#pragma unroll
    for (int l = 0; l < 3; ++l) {
      const float* Wl = Ws[l];
      const float* gl = Gs[l];
      const float* bl = Bs[l];
      const int Kd = Ks[l], Co = Cos[l];
      float* inBuf = (l == 1) ? bufB : bufA;
      float* outBuf = (l == 0) ? bufB : bufA;
      const int col = lane & 15;
      const int khalf = (lane >> 4) << 1;  // lanes 0-15: K 0,1 ; lanes 16-31: K 2,3
      const int ntile = Co >> 4;
      for (int t = wave; t < ntile; t += 4) {
        const int row = (t << 4) + col;
        const float* wrow = Wl + (size_t)row * Kd;
        __builtin_prefetch(wrow, 0, 1);
        v8f acc = {0.f, 0.f, 0.f, 0.f, 0.f, 0.f, 0.f, 0.f};
#pragma unroll 2
        for (int k0 = 0; k0 < Kd; k0 += 4) {
          int kk = k0 + khalf;
          v2f a = *(const v2f*)(wrow + kk);  // 8B-aligned: stride mult of 4, kk even
          v2f bf;
          bf.x = inBuf[kk * 16 + col];
          bf.y = inBuf[(kk + 1) * 16 + col];
          acc = __builtin_amdgcn_wmma_f32_16x16x4_f32(false, a, false, bf,
                                                      (short)0, acc, false, false);
        }
        const int ob = (t << 4) + ((lane >> 4) << 3);  // D: vgpr r -> M=r (+8 upper half)
#pragma unroll
        for (int r = 0; r < 8; ++r) {
          int o = ob + r;
          float val = fmaxf(acc[r] * gl[o] + bl[o], 0.0f);
          if (l < 2) {
            outBuf[o * 16 + col] = val;
          } else {
            // column-max over the 16 neighbor columns (within each lane half)
#pragma unroll
            for (int moff = 1; moff < 16; moff <<= 1)
              val = fmaxf(val, __shfl_xor(val, moff, 32));
            if (col == 0) maxbuf[o] = fmaxf(maxbuf[o], val);
          }
        }
      }
      __syncthreads();
    }
  }

  for (int c = tid; c < Co2; c += 128)
    fout[((size_t)b * CoutTotal + oOff + c) * S + s] = maxbuf[c];
}

// ---------------------------------------------------------------------------
// Small FC heads (B=8 rows -> plain VALU)
// ---------------------------------------------------------------------------
__global__ void fc_relu_kernel(const float* __restrict__ x, const float* __restrict__ W,
                               const float* __restrict__ g, const float* __restrict__ bias,
                               float* __restrict__ out, int K, int Co) {
  int t = blockIdx.x * blockDim.x + threadIdx.x;
  if (t >= BPC * Co) return;
  int b = t / Co, o = t % Co;
  const float* xb = x + (size_t)b * K;
  const float* wr = W + (size_t)o * K;
  float acc = 0.0f;
  for (int k = 0; k < K; ++k) acc += xb[k] * wr[k];
  out[t] = fmaxf(acc * g[o] + bias[o], 0.0f);
}

__global__ void fc_kernel(const float* __restrict__ x, const float* __restrict__ W,
                          float* __restrict__ out, int K, int Co) {
  int t = blockIdx.x * blockDim.x + threadIdx.x;
  if (t >= BPC * Co) return;
  int b = t / Co, o = t % Co;
  const float* xb = x + (size_t)b * K;
  const float* wr = W + (size_t)o * K;
  float acc = 0.0f;
  for (int k = 0; k < K; ++k) acc += xb[k] * wr[k];
  out[t] = acc;
}

// ---------------------------------------------------------------------------
// Host orchestration
// ---------------------------------------------------------------------------
static inline size_t smemSA(int Cin, int Co0, int Co1, int Co2) {
  int CinP = (Cin + 3) & ~3;
  int szA = ((CinP > Co1) ? CinP : Co1) * 16;
  int szB = Co0 * 16;
  return (size_t)(szA + szB + Co2) * sizeof(float);
}

extern "C" void kernel_launch(void* const* d_in, const int* in_sizes, int n_in,
                              void* d_out, int out_size, void* d_ws, size_t ws_size,
                              hipStream_t stream) {
  (void)in_sizes; (void)n_in; (void)out_size; (void)ws_size;
  const int B = 8, N = 16384, S1 = 512, S2 = 128;

  const float* P[70];
  for (int i = 0; i < 70; ++i) P[i] = (const float*)d_in[i];
  const float* pc  = P[0];
  const float* ctr = P[1];
  const float* szr = P[2];

  char* ws = (char*)d_ws;
  size_t off = 0;
  auto alloc = [&](size_t bytes) -> void* {
    void* p = ws + off;
    off = (off + bytes + 255) & ~(size_t)255;
    return p;
  };
  float* xyz0   = (float*)alloc((size_t)B * N * 3 * 4);
  float* feats0 = (float*)alloc((size_t)B * 4 * N * 4);
  int*   fps1   = (int*)alloc((size_t)B * S1 * 4);
  float* nx1    = (float*)alloc((size_t)B * S1 * 3 * 4);
  int*   ball1_0 = (int*)alloc((size_t)B * S1 * 16 * 4);
  int*   ball1_1 = (int*)alloc((size_t)B * S1 * 32 * 4);
  int*   ball1_2 = (int*)alloc((size_t)B * S1 * 128 * 4);
  float* feats1 = (float*)alloc((size_t)B * 320 * S1 * 4);
  int*   fps2   = (int*)alloc((size_t)B * S2 * 4);
  float* nx2    = (float*)alloc((size_t)B * S2 * 3 * 4);
  int*   ball2_0 = (int*)alloc((size_t)B * S2 * 32 * 4);
  int*   ball2_1 = (int*)alloc((size_t)B * S2 * 64 * 4);
  int*   ball2_2 = (int*)alloc((size_t)B * S2 * 128 * 4);
  float* feats2 = (float*)alloc((size_t)B * 640 * S2 * 4);
  float* x3     = (float*)alloc((size_t)B * 1024 * 4);
  float* h      = (float*)alloc((size_t)B * 512 * 4);

  // padded layer-0 weights: SA1 (Ci=7->8), SA2 (323->324), SA3 (643->644)
  const int co1[3][3] = {{32, 32, 64}, {64, 64, 128}, {64, 96, 128}};
  const int co2[3][3] = {{64, 64, 128}, {128, 128, 256}, {128, 128, 256}};
  float* w0p1[3], *w0p2[3], *w0p3;
  for (int i = 0; i < 3; ++i) w0p1[i] = (float*)alloc((size_t)co1[i][0] * 8 * 4);
  for (int i = 0; i < 3; ++i) w0p2[i] = (float*)alloc((size_t)co2[i][0] * 324 * 4);
  w0p3 = (float*)alloc((size_t)256 * 644 * 4);

  for (int i = 0; i < 3; ++i) {
    int n = co1[i][0] * 8;
    pad_w_kernel<<<(n + 255) / 256, 256, 0, stream>>>(P[3 + i * 9], w0p1[i], co1[i][0], 7, 8);
  }
  for (int i = 0; i < 3; ++i) {
    int n = co2[i][0] * 324;
    pad_w_kernel<<<(n + 255) / 256, 256, 0, stream>>>(P[30 + i * 9], w0p2[i], co2[i][0], 323, 324);
  }
  {
    int n = 256 * 644;
    pad_w_kernel<<<(n + 255) / 256, 256, 0, stream>>>(P[57], w0p3, 256, 643, 644);
  }

  // ---- inputs ----
  build_inputs_kernel<<<(B * N + 255) / 256, 256, 0, stream>>>(pc, ctr, szr, xyz0, feats0, N);

  // ---- SA1 ----
  fps_kernel<16><<<B, 1024, 0, stream>>>(xyz0, fps1, N, S1);
  gather_xyz_kernel<<<(B * S1 * 3 + 255) / 256, 256, 0, stream>>>(xyz0, fps1, nx1, N, S1);

  int*  ball1[3] = {ball1_0, ball1_1, ball1_2};
  float rad1[3] = {0.1f, 0.2f, 0.4f};
  int   nss1[3] = {16, 32, 128};
  for (int i = 0; i < 3; ++i)
    ball_query_kernel<<<(B * S1 + 7) / 8, 256, 0, stream>>>(xyz0, nx1, ball1[i], N, S1,
                                                            nss1[i], rad1[i] * rad1[i]);
  const int off1[3] = {0, 64, 192};
  for (int i = 0; i < 3; ++i) {
    const float* const* q = P + 3 + i * 9;
    sa_branch_kernel<<<B * S1, 128, smemSA(7, co1[i][0], co1[i][1], co1[i][2]), stream>>>(
        xyz0, feats0, nx1, ball1[i], feats1,
        N, 4, S1, nss1[i], off1[i], 320,
        w0p1[i], q[1], q[2], co1[i][0],
        q[3], q[4], q[5], co1[i][1],
        q[6], q[7], q[8], co1[i][2]);
  }

  // ---- SA2 ----
  fps_kernel<1><<<B, 512, 0, stream>>>(nx1, fps2, S1, S2);
  gather_xyz_kernel<<<(B * S2 * 3 + 255) / 256, 256, 0, stream>>>(nx1, fps2, nx2, S1, S2);

  int*  ball2[3] = {ball2_0, ball2_1, ball2_2};
  float rad2[3] = {0.2f, 0.4f, 0.8f};
  int   nss2[3] = {32, 64, 128};
  for (int i = 0; i < 3; ++i)
    ball_query_kernel<<<(B * S2 + 7) / 8, 256, 0, stream>>>(nx1, nx2, ball2[i], S1, S2,
                                                            nss2[i], rad2[i] * rad2[i]);
  const int off2[3] = {0, 128, 384};
  for (int i = 0; i < 3; ++i) {
    const float* const* q = P + 30 + i * 9;
    sa_branch_kernel<<<B * S2, 128, smemSA(323, co2[i][0], co2[i][1], co2[i][2]), stream>>>(
        nx1, feats1, nx2, ball2[i], feats2,
        S1, 320, S2, nss2[i], off2[i], 640,
        w0p2[i], q[1], q[2], co2[i][0],
        q[3], q[4], q[5], co2[i][1],
        q[6], q[7], q[8], co2[i][2]);
  }

  // ---- SA3 (global, identity grouping, no centroid subtract) ----
  {
    const float* const* q = P + 57;
    sa_branch_kernel<<<B, 128, smemSA(643, 256, 512, 1024), stream>>>(
        nx2, feats2, nullptr, nullptr, x3,
        S2, 640, 1, 128, 0, 1024,
        w0p3, q[1], q[2], 256,
        q[3], q[4], q[5], 512,
        q[6], q[7], q[8], 1024);
  }

  // ---- FC head ----
  fc_relu_kernel<<<(B * 512 + 255) / 256, 256, 0, stream>>>(x3, P[66], P[67], P[68], h, 1024, 512);
  fc_kernel<<<(B * 256 + 255) / 256, 256, 0, stream>>>(h, P[69], (float*)d_out, 512, 256);
}